// GNMT_Decoder_Layer_50010599195161
// MI455X (gfx1250) — compile-verified
//
#include <hip/hip_runtime.h>
#include <math.h>

// GNMT decoder layer for MI455X (gfx1250, wave32, WMMA f32 16x16x4).
// B=8, S=64, H=512, 4H=2048, NRES=2.

typedef __attribute__((ext_vector_type(2))) float v2f;
typedef __attribute__((ext_vector_type(8))) float v8f;

__device__ __forceinline__ float sigm(float x) { return 1.0f / (1.0f + expf(-x)); }

// ---------------------------------------------------------------------------
// GEMM: out[m][n] = sum_k A[m][k] * W[n][k]  (+ b0[n] + b1[n])
// A is [M=512][512] (optionally two concatenated 512-wide chunks A0,A1 whose
// W columns are [0,512) and [512,1024)); W has row stride ldw.
// One 16x16 output tile per wave via V_WMMA_F32_16X16X4_F32.
// Fragment layout (ISA 7.12.2, 32-bit A 16x4): lane = m + 16*(k>>1), vgpr=k&1.
// ---------------------------------------------------------------------------
__global__ __launch_bounds__(256)
void gemm_xwT(const float* __restrict__ A0, const float* __restrict__ A1,
              const float* __restrict__ W, int ldw,
              const float* __restrict__ b0, const float* __restrict__ b1,
              float* __restrict__ out, int N)
{
    const int lane  = threadIdx.x & 31;
    const int wave  = threadIdx.x >> 5;
    const int ntile = blockIdx.x * 8 + wave;   // N/16 tiles
    const int mtile = blockIdx.y;              // M/16 tiles (M=512 -> 32)
    const int m   = lane & 15;                 // A row within tile / W col within tile
    const int kk  = (lane >> 4) << 1;          // 0 or 2
    const size_t ncol = (size_t)ntile * 16 + m;

    v8f acc = {0.f,0.f,0.f,0.f,0.f,0.f,0.f,0.f};

    {
        const float* arow = A0 + ((size_t)mtile * 16 + m) * 512;
        const float* wrow = W + ncol * (size_t)ldw;
        #pragma unroll 4
        for (int k0 = 0; k0 < 512; k0 += 4) {
            v2f a = *(const v2f*)(arow + k0 + kk);
            v2f b = *(const v2f*)(wrow + k0 + kk);
            acc = __builtin_amdgcn_wmma_f32_16x16x4_f32(false, a, false, b,
                                                        (short)0, acc, false, false);
        }
    }
    if (A1) {
        const float* arow = A1 + ((size_t)mtile * 16 + m) * 512;
        const float* wrow = W + ncol * (size_t)ldw + 512;
        #pragma unroll 4
        for (int k0 = 0; k0 < 512; k0 += 4) {
            v2f a = *(const v2f*)(arow + k0 + kk);
            v2f b = *(const v2f*)(wrow + k0 + kk);
            acc = __builtin_amdgcn_wmma_f32_16x16x4_f32(false, a, false, b,
                                                        (short)0, acc, false, false);
        }
    }

    float bias = 0.f;
    if (b0) bias += b0[ncol];
    if (b1) bias += b1[ncol];

    const int rbase = (lane >> 4) << 3;        // C/D: vgpr i -> rows i and i+8
    #pragma unroll
    for (int i = 0; i < 8; ++i) {
        int row = i + rbase;
        out[((size_t)mtile * 16 + row) * N + ncol] = acc[i] + bias;
    }
}

// ---------------------------------------------------------------------------
// One-shot swizzle of Whh (2048 x 512, row-major) into WMMA B-fragment order:
//   swz float2 index  c2 = (k0/4)*4096 + tile*32 + lane
//   holds Whh[tile*16 + (lane&15)][k0 + 2*(lane>>4) + {0,1}]
// so the scan's B-fragment load is a fully coalesced contiguous 256B wave read.
// ---------------------------------------------------------------------------
__global__ __launch_bounds__(256)
void swz_whh(const float* __restrict__ Whh, float* __restrict__ swz)
{
    const int c2   = blockIdx.x * 256 + threadIdx.x;  // 0 .. 524287 (float2 units)
    const int lane = c2 & 31;
    const int tl   = (c2 >> 5) & 127;
    const int kIt  = c2 >> 12;
    const int n = tl * 16 + (lane & 15);
    const int k = kIt * 4 + ((lane >> 4) << 1);
    v2f v = *(const v2f*)(Whh + (size_t)n * 512 + k);
    *(v2f*)(swz + (size_t)c2 * 2) = v;
}

// ---------------------------------------------------------------------------
// Attention (fully parallel over (b,t)): one block per (b,t).
// scores[s] = sum_h tanh(proj_v[b,s,h] + qW[b,t,h]) * V[h] + mask
// ctx[b,t,:] = softmax(scores) @ enc[b,:,:]
// ---------------------------------------------------------------------------
__global__ __launch_bounds__(256)
void attn_kernel(const float* __restrict__ proj_v, const float* __restrict__ qW,
                 const float* __restrict__ V, const float* __restrict__ enc,
                 const int* __restrict__ len, float* __restrict__ ctx)
{
    __shared__ float qv[512];
    __shared__ float vv[512];
    __shared__ float sc[64];
    __shared__ float wts[64];

    const int bt = blockIdx.x;          // 0..511
    const int b = bt >> 6;
    const int tid = threadIdx.x;
    const int lane = tid & 31, wave = tid >> 5;

    qv[tid]       = qW[(size_t)bt * 512 + tid];
    qv[tid + 256] = qW[(size_t)bt * 512 + tid + 256];
    vv[tid]       = V[tid];
    vv[tid + 256] = V[tid + 256];
    __syncthreads();

    const int L = len[b];
    #pragma unroll
    for (int j = 0; j < 8; ++j) {
        int s = wave + j * 8;
        const float* pv = proj_v + ((size_t)b * 64 + s) * 512;
        float p = 0.f;
        for (int h = lane; h < 512; h += 32)
            p += tanhf(pv[h] + qv[h]) * vv[h];
        for (int off = 16; off; off >>= 1) p += __shfl_xor(p, off, 32);
        if (lane == 0) sc[s] = p + (s < L ? 0.0f : -1.0e9f);
    }
    __syncthreads();

    if (tid < 32) {
        float s0 = sc[tid], s1 = sc[tid + 32];
        float mx = fmaxf(s0, s1);
        for (int off = 16; off; off >>= 1) mx = fmaxf(mx, __shfl_xor(mx, off, 32));
        float e0 = expf(s0 - mx), e1 = expf(s1 - mx);
        float sum = e0 + e1;
        for (int off = 16; off; off >>= 1) sum += __shfl_xor(sum, off, 32);
        wts[tid] = e0 / sum;
        wts[tid + 32] = e1 / sum;
    }
    __syncthreads();

    for (int h = tid; h < 512; h += 256) {
        const float* e = enc + ((size_t)b * 64) * 512 + h;
        float a = 0.f;
        #pragma unroll 8
        for (int s = 0; s < 64; ++s) a += wts[s] * e[(size_t)s * 512];
        ctx[(size_t)bt * 512 + h] = a;
    }
}

// ---------------------------------------------------------------------------
// LSTM scan: single workgroup (512 thr = 16 waves), 64 sequential steps.
// Per step: gates = xIh[:,t,:] + mm @ Whh^T   (mm padded to 16 rows, K=512,
// N=2048), via WMMA f32 16x16x4 with the pre-swizzled Whh (coalesced B loads).
// Wave w owns j-tiles {w, w+16}; for each it computes all four gate tiles
// (i: n=j, f: 512+j, g: 1024+j, o: 1536+j), so the LSTM pointwise update runs
// in-register off the WMMA accumulators. swap!=0 reproduces the reference's
// swapped-carry layer-1 cell.
// ---------------------------------------------------------------------------
__global__ __launch_bounds__(512)
void lstm_scan(const float* __restrict__ xIh, const float* __restrict__ whhSwz,
               const float* __restrict__ initMM, const float* __restrict__ initCC,
               const float* __restrict__ addTo, const int* __restrict__ len,
               float* __restrict__ out, int swap)
{
    __shared__ float mm[16 * 513];   // state multiplied by Whh; rows 8..15 zero; +1 pad vs banks
    __shared__ float cc[8 * 512];    // cell-side state

    const int tid  = threadIdx.x;
    const int lane = tid & 31;
    const int wave = tid >> 5;       // 0..15

    for (int i = tid; i < 16 * 513; i += 512) mm[i] = 0.0f;
    __syncthreads();
    for (int e = tid; e < 8 * 512; e += 512) {
        int b = e >> 9, j = e & 511;
        mm[b * 513 + j] = initMM ? initMM[e] : 0.0f;
        cc[e]           = initCC ? initCC[e] : 0.0f;
    }
    __syncthreads();

    const int m     = lane & 15;
    const int kk    = (lane >> 4) << 1;
    const int rbase = (lane >> 4) << 3;
    const int jt[2] = { wave, wave + 16 };

    int lenv[8];
    #pragma unroll
    for (int i = 0; i < 8; ++i) lenv[i] = len ? len[i] : 64;

    // per-(p,g) byte-free float offsets into one k-slab of the swizzled Whh
    size_t btile[2][4];
    #pragma unroll
    for (int p = 0; p < 2; ++p)
        #pragma unroll
        for (int g = 0; g < 4; ++g)
            btile[p][g] = (size_t)(g * 32 + jt[p]) * 64 + lane * 2;

    for (int t = 0; t < 64; ++t) {
        v8f acc[2][4];
        const v8f z = {0.f,0.f,0.f,0.f,0.f,0.f,0.f,0.f};
        #pragma unroll
        for (int p = 0; p < 2; ++p)
            #pragma unroll
            for (int g = 0; g < 4; ++g) acc[p][g] = z;

        #pragma unroll 2
        for (int k0 = 0; k0 < 512; k0 += 4) {
            v2f a;
            a.x = mm[m * 513 + k0 + kk];
            a.y = mm[m * 513 + k0 + kk + 1];
            const float* slab = whhSwz + ((size_t)(k0 >> 2)) * 8192;
            #pragma unroll
            for (int p = 0; p < 2; ++p)
                #pragma unroll
                for (int g = 0; g < 4; ++g) {
                    v2f bf = *(const v2f*)(slab + btile[p][g]);
                    acc[p][g] = __builtin_amdgcn_wmma_f32_16x16x4_f32(
                        false, a, false, bf, (short)0, acc[p][g], false, false);
                }
        }
        __syncthreads();   // everyone done reading mm before it is overwritten

        #pragma unroll
        for (int p = 0; p < 2; ++p) {
            const int j = jt[p] * 16 + m;   // D col = lane&15
            #pragma unroll
            for (int i = 0; i < 8; ++i) {
                const int row = i + rbase;  // D: vgpr i -> rows i / i+8
                if (row < 8) {
                    const size_t xb = ((size_t)(row * 64 + t)) * 2048 + j;
                    float gi = acc[p][0][i] + xIh[xb];
                    float gf = acc[p][1][i] + xIh[xb + 512];
                    float gg = acc[p][2][i] + xIh[xb + 1024];
                    float go = acc[p][3][i] + xIh[xb + 1536];
                    float cprev = cc[row * 512 + j];
                    float celln = sigm(gf) * cprev + sigm(gi) * tanhf(gg);
                    float hn    = sigm(go) * tanhf(celln);
                    float val = (t < lenv[row]) ? hn : 0.0f;
                    const size_t ob = ((size_t)(row * 64 + t)) * 512 + j;
                    if (addTo) val += addTo[ob];
                    out[ob] = val;
                    if (swap) { mm[row * 513 + j] = celln; cc[row * 512 + j] = hn; }
                    else      { mm[row * 513 + j] = hn;    cc[row * 512 + j] = celln; }
                }
            }
        }
        __syncthreads();   // writes to mm/cc visible before next step's matvec
    }
}

// ---------------------------------------------------------------------------
extern "C" void kernel_launch(void* const* d_in, const int* in_sizes, int n_in,
                              void* d_out, int out_size, void* d_ws, size_t ws_size,
                              hipStream_t stream)
{
    const float* enc   = (const float*)d_in[0];
    const float* bh    = (const float*)d_in[1];   // backward_hidden_state
    const float* bc    = (const float*)d_in[2];   // backward_cell_state
    const float* Ww    = (const float*)d_in[3];   // (512, 1024): Wq=[:, :512], Wv=[:, 512:]
    const float* Wb    = (const float*)d_in[4];
    const float* Vv    = (const float*)d_in[5];
    const float* l1Wih = (const float*)d_in[6];
    const float* l1Whh = (const float*)d_in[7];
    const float* l1bih = (const float*)d_in[8];
    const float* l1bhh = (const float*)d_in[9];
    const float* l2Wih = (const float*)d_in[10];
    const float* l2Whh = (const float*)d_in[11];
    const float* l2bih = (const float*)d_in[12];
    const float* l2bhh = (const float*)d_in[13];
    const float* rWih  = (const float*)d_in[14];  // (2, 2048, 1024)
    const float* rWhh  = (const float*)d_in[15];  // (2, 2048, 512)
    const float* rbih  = (const float*)d_in[16];  // (2, 2048)
    const float* rbhh  = (const float*)d_in[17];
    const int*   ilen  = (const int*)d_in[18];
    float* out = (float*)d_out;

    float* ws     = (float*)d_ws;
    float* proj_v = ws;                        // 512*512
    float* qW     = proj_v + 512 * 512;        // 512*512
    float* ctx    = qW     + 512 * 512;        // 512*512
    float* h1     = ctx    + 512 * 512;        // 512*512
    float* xIh    = h1     + 512 * 512;        // 512*2048
    float* hidA   = xIh    + 512 * 2048;       // 512*512
    float* hidB   = hidA   + 512 * 512;        // 512*512
    float* swzW   = hidB   + 512 * 512;        // 2048*512 (reused per layer) ~14 MB total

    const dim3 blk(256);
    const dim3 swzGrid(2048);

    // Parallel precompute: proj_v = enc @ Wv^T + Wb ;  qW = enc @ Wq^T
    gemm_xwT<<<dim3(4, 32), blk, 0, stream>>>(enc, nullptr, Ww + 512, 1024, Wb, nullptr, proj_v, 512);
    gemm_xwT<<<dim3(4, 32), blk, 0, stream>>>(enc, nullptr, Ww,       1024, nullptr, nullptr, qW, 512);

    // Attention is carry-independent -> all (b,t) in parallel.
    attn_kernel<<<dim3(512), blk, 0, stream>>>(proj_v, qW, Vv, enc, ilen, ctx);

    // Layer 1: xIh = [enc, ctx] @ l1_Wih^T + biases; scan with swapped carry
    // (reference calls _lstm_cell(x, c, h): Whh multiplies carry_c, cell side is carry_h).
    gemm_xwT<<<dim3(16, 32), blk, 0, stream>>>(enc, ctx, l1Wih, 1024, l1bih, l1bhh, xIh, 2048);
    swz_whh<<<swzGrid, blk, 0, stream>>>(l1Whh, swzW);
    lstm_scan<<<dim3(1), dim3(512), 0, stream>>>(xIh, swzW, bc, bh, nullptr, nullptr, h1, 1);

    // Layer 2: x = [h1, ctx]; output masked by valid.
    gemm_xwT<<<dim3(16, 32), blk, 0, stream>>>(h1, ctx, l2Wih, 1024, l2bih, l2bhh, xIh, 2048);
    swz_whh<<<swzGrid, blk, 0, stream>>>(l2Whh, swzW);
    lstm_scan<<<dim3(1), dim3(512), 0, stream>>>(xIh, swzW, nullptr, nullptr, nullptr, ilen, hidA, 0);

    // Residual layer 0: hidB = hidA + mask * scan([hidA, ctx])
    gemm_xwT<<<dim3(16, 32), blk, 0, stream>>>(hidA, ctx, rWih, 1024, rbih, rbhh, xIh, 2048);
    swz_whh<<<swzGrid, blk, 0, stream>>>(rWhh, swzW);
    lstm_scan<<<dim3(1), dim3(512), 0, stream>>>(xIh, swzW, nullptr, nullptr, hidA, ilen, hidB, 0);

    // Residual layer 1: out = hidB + mask * scan([hidB, ctx])
    gemm_xwT<<<dim3(16, 32), blk, 0, stream>>>(hidB, ctx, rWih + 2048 * 1024, 1024,
                                               rbih + 2048, rbhh + 2048, xIh, 2048);
    swz_whh<<<swzGrid, blk, 0, stream>>>(rWhh + 2048 * 512, swzW);
    lstm_scan<<<dim3(1), dim3(512), 0, stream>>>(xIh, swzW, nullptr, nullptr,
                                                 hidB, ilen, out, 0);
}